// BeamSearch_11295763988573
// MI455X (gfx1250) — compile-verified
//
#include <hip/hip_runtime.h>

#define B_ 64
#define N_ 1024
#define TILE_I 16

typedef unsigned int v4u __attribute__((ext_vector_type(4)));
typedef int          v4i __attribute__((ext_vector_type(4)));
typedef int          v8i __attribute__((ext_vector_type(8)));

// ---------------------------------------------------------------------------
// Kernel 1: per-(b,j) preprocessing.
//   new_mask[b,j] = m * c1 * c2   (m = mask with pres_act column zeroed)
//   fpresent[b,j] = (arr + max(0, open-arr)) + dur     (exact ref op order)
// Also stages contiguous copies of open/close/dur/new_mask/dlast into ws so
// the adj kernel gets fully coalesced, aligned loads.
// ---------------------------------------------------------------------------
__global__ void prep_kernel(const float* __restrict__ inputs,
                            const float* __restrict__ dist,
                            const float* __restrict__ mask,
                            const float* __restrict__ ptime,
                            const int*   __restrict__ pres_act,
                            float* __restrict__ out_mask,   // d_out + 1
                            float* __restrict__ ws_fp,
                            float* __restrict__ ws_op,
                            float* __restrict__ ws_cl,
                            float* __restrict__ ws_du,
                            float* __restrict__ ws_val,
                            float* __restrict__ ws_dlast) {
    const int b = blockIdx.y;
    const int j = blockIdx.x * blockDim.x + threadIdx.x;

    const int   pa = pres_act[b];
    const float pt = ptime[b];

    const float* row = inputs + ((size_t)b * N_ + j) * 4;
    const float open  = row[0];
    const float close = row[1];
    const float dur   = row[2];
    const float T0    = inputs[3];                    // inputs[0,0,ARR]

    const float dl  = dist[(size_t)j * N_ + (N_ - 1)];   // dlast[j]
    const float arr = dist[(size_t)pa * N_ + j] + pt;
    const float w   = fmaxf(0.0f, open - arr);
    const float t   = arr + w;                        // arrive + wait (exact)

    float m = mask[(size_t)b * N_ + j];
    if (j == pa) m = 0.0f;
    const bool c1 = (t <= close);
    const bool c2 = (((t + dur) + dl) <= T0);
    const float nm = (c1 && c2) ? m : 0.0f;

    const size_t idx = (size_t)b * N_ + j;
    out_mask[idx] = nm;                               // final output: keep RT
    ws_val[idx]   = nm;
    ws_fp[idx]    = t + dur;      // fpresent (farrive==arrive, same expr)
    ws_op[idx]    = open;
    ws_cl[idx]    = close;
    ws_du[idx]    = dur;
    if (b == 0) ws_dlast[j] = dl;
}

// ---------------------------------------------------------------------------
// Kernel 2: tiny scalar outputs: done, future_actions (as float),
// present_time_new, step_mask.  One block of 64 threads (2 waves).
// ---------------------------------------------------------------------------
__global__ void finalize_kernel(const float* __restrict__ inputs,
                                const float* __restrict__ dist,
                                const float* __restrict__ ptime,
                                const int*   __restrict__ pres_act,
                                const int*   __restrict__ fut,
                                const float* __restrict__ ws_val,
                                float* __restrict__ out) {
    __shared__ int anyflag;
    const int t = threadIdx.x;
    if (t == 0) anyflag = 0;
    __syncthreads();

    const size_t OFF_FUT  = (size_t)1 + (size_t)B_ * N_ + (size_t)B_ * N_ * N_;
    const size_t OFF_PTN  = OFF_FUT + B_;
    const size_t OFF_STEP = OFF_PTN + B_;

    if (t < B_) {
        if (ws_val[(size_t)t * N_ + (N_ - 1)] > 0.0f) atomicOr(&anyflag, 1);

        const int   pa = pres_act[t];
        const int   fa = fut[t];
        const float pt = ptime[t];
        const float* row = inputs + ((size_t)t * N_ + fa) * 4;
        const float arr = dist[(size_t)pa * N_ + fa] + pt;
        const float w   = fmaxf(0.0f, row[0] - arr);
        const float ptn = (arr + w) + row[2];

        out[OFF_FUT  + t] = (float)fa;
        out[OFF_PTN  + t] = ptn;
        out[OFF_STEP + t] = 1.0f;
    }
    __syncthreads();
    if (t == 0) out[0] = anyflag ? 0.0f : 1.0f;   // done = !any(...)
}

// ---------------------------------------------------------------------------
// Kernel 3: adj[b,i,j].  Block = (i-tile of 16 rows, batch b), 256 threads.
// Wave 0 issues a TDM tensor_load_to_lds of the 16x1024 dist_mat tile
// (64 KB) into LDS, waits on TENSORcnt; per-j scalars overlap via normal
// vector loads.  Each thread owns 4 j-columns, sweeps 16 rows, and emits
// non-temporal streaming stores (adj >> L2, never re-read).
// ---------------------------------------------------------------------------
__global__ __launch_bounds__(256)
void adj_kernel(const float* __restrict__ dist,
                const float* __restrict__ inputs,
                const float* __restrict__ ws_fp,
                const float* __restrict__ ws_op,
                const float* __restrict__ ws_cl,
                const float* __restrict__ ws_du,
                const float* __restrict__ ws_val,
                const float* __restrict__ ws_dlast,
                float* __restrict__ adj) {           // d_out + 1 + B*N
    __shared__ float lds_dist[TILE_I * N_];          // 64 KB, sole LDS object

    const int b   = blockIdx.y;
    const int i0  = blockIdx.x * TILE_I;
    const int tid = threadIdx.x;

    // ---- TDM: async DMA of dist_mat[i0 : i0+16, :] into LDS (wave 0) ----
    if (tid < 32) {
        // Generic->LDS offset: LDS aperture keeps the offset in addr[31:0].
        unsigned lds_base = (unsigned)(unsigned long long)(void*)&lds_dist[0];
        unsigned long long ga =
            (unsigned long long)(const void*)(dist + (size_t)i0 * N_);

        v4u g0;
        g0[0] = 1u;                                   // count=1, user mode
        g0[1] = lds_base;                             // lds_addr (bytes)
        g0[2] = (unsigned)ga;                         // global_addr[31:0]
        g0[3] = (unsigned)(ga >> 32) | (2u << 30);    // addr[56:32] | type=2

        v8i g1;
        g1[0] = (int)(2u << 16);                      // data_size=2 (4 bytes)
        g1[1] = (int)(((unsigned)N_ & 0xFFFFu) << 16);         // tensor_dim0 lo
        g1[2] = (int)((((unsigned)N_ >> 16) & 0xFFFFu) |
                      (((unsigned)N_ & 0xFFFFu) << 16));       // d0 hi | d1 lo
        g1[3] = (int)((((unsigned)N_ >> 16) & 0xFFFFu) |
                      (((unsigned)N_ & 0xFFFFu) << 16));       // d1 hi | tile_dim0
        g1[4] = TILE_I;                               // tile_dim1 (tile_dim2=0)
        g1[5] = N_;                                   // tensor_dim0_stride lo
        g1[6] = 0;
        g1[7] = 0;

        v4i z4 = {0, 0, 0, 0};                        // groups 2/3 unused (2-D)
        v8i z8 = {0, 0, 0, 0, 0, 0, 0, 0};            // extra group (clang-23 form)
        __builtin_amdgcn_tensor_load_to_lds(g0, g1, z4, z4, z8, 0);
        __builtin_amdgcn_s_wait_tensorcnt(0);
    }

    // ---- per-j column state (overlaps with the TDM transfer) ----
    float cl4[4], du4[4], dl4[4], va4[4];
#pragma unroll
    for (int k = 0; k < 4; ++k) {
        const int j = tid + k * 256;
        const size_t idx = (size_t)b * N_ + j;
        cl4[k] = ws_cl[idx];
        du4[k] = ws_du[idx];
        va4[k] = ws_val[idx];
        dl4[k] = ws_dlast[j];
    }
    const float Tb = inputs[(size_t)b * N_ * 4 + 3];  // inputs[b,0,ARR]

    __syncthreads();                                  // LDS tile ready

    float* out_row0 = adj + ((size_t)b * N_ + i0) * N_;
#pragma unroll 4
    for (int r = 0; r < TILE_I; ++r) {
        const int i = i0 + r;
        const float fp = ws_fp[(size_t)b * N_ + i];   // uniform per row
        const float op = ws_op[(size_t)b * N_ + i];
        float* orow = out_row0 + (size_t)r * N_;
#pragma unroll
        for (int k = 0; k < 4; ++k) {
            const int j = tid + k * 256;
            const float d   = lds_dist[r * N_ + j];
            const float arr = d + fp;
            const float w   = fmaxf(0.0f, op - arr);
            const float t2  = arr + w;                // arr2 + wait2 (exact)
            const bool ok = (t2 <= cl4[k]) &&
                            (((t2 + du4[k]) + dl4[k]) <= Tb);
            float o = ok ? va4[k] : 0.0f;
            if (i == j) o = 1.0f;                     // eye -> 1.0
            __builtin_nontemporal_store(o, &orow[j]); // streaming store (NT)
        }
    }
}

// ---------------------------------------------------------------------------
extern "C" void kernel_launch(void* const* d_in, const int* in_sizes, int n_in,
                              void* d_out, int out_size, void* d_ws, size_t ws_size,
                              hipStream_t stream) {
    const float* inputs   = (const float*)d_in[0];   // (B,N,4)
    const float* dist     = (const float*)d_in[1];   // (N,N)
    const float* mask     = (const float*)d_in[2];   // (B,N)
    const float* ptime    = (const float*)d_in[3];   // (B,1)
    const int*   pres_act = (const int*)d_in[4];     // (B,)
    const int*   fut      = (const int*)d_in[5];     // (B,)

    float* out = (float*)d_out;
    float* ws  = (float*)d_ws;

    const size_t BN = (size_t)B_ * N_;
    float* ws_fp    = ws;
    float* ws_op    = ws + BN;
    float* ws_cl    = ws + 2 * BN;
    float* ws_du    = ws + 3 * BN;
    float* ws_val   = ws + 4 * BN;
    float* ws_dlast = ws + 5 * BN;                   // N floats

    // d_out layout: [done(1)][new_mask(BN)][adj(B*N*N)][fut(B)][ptn(B)][step(B)]
    float* out_mask = out + 1;
    float* out_adj  = out + 1 + BN;

    prep_kernel<<<dim3(N_ / 256, B_), 256, 0, stream>>>(
        inputs, dist, mask, ptime, pres_act,
        out_mask, ws_fp, ws_op, ws_cl, ws_du, ws_val, ws_dlast);

    finalize_kernel<<<1, 64, 0, stream>>>(
        inputs, dist, ptime, pres_act, fut, ws_val, out);

    adj_kernel<<<dim3(N_ / TILE_I, B_), 256, 0, stream>>>(
        dist, inputs, ws_fp, ws_op, ws_cl, ws_du, ws_val, ws_dlast, out_adj);
}